// ChamferDistance_62723702391632
// MI455X (gfx1250) — compile-verified
//
#include <hip/hip_runtime.h>

typedef float v2f __attribute__((ext_vector_type(2)));
typedef float v8f __attribute__((ext_vector_type(8)));

#define N_PTS            8192
#define BATCHES          4
#define CHUNK            2048
#define TILES_PER_CHUNK  (CHUNK / 16)
#define WAVES_PER_WG     8
#define ROWS_PER_WG      (WAVES_PER_WG * 16)

__global__ void chamfer_init_kernel(float* out) { out[0] = 0.0f; }

// One direction of Chamfer: for each point in P, min squared distance to any
// point in Q; per-row mins are atomically accumulated (pre-scaled) into out[0].
__global__ __launch_bounds__(256)
void chamfer_dir_kernel(const float* __restrict__ P, const float* __restrict__ Q,
                        float* __restrict__ out, float scale)
{
    __shared__ float4 sB[CHUNK];  // (-2x, -2y, -2z, 1) per target point
    __shared__ float  sN[CHUNK];  // |q|^2 per target point

    const int blocks_per_batch = N_PTS / ROWS_PER_WG;
    const int b      = blockIdx.x / blocks_per_batch;
    const int rowblk = blockIdx.x % blocks_per_batch;
    const int tid    = threadIdx.x;
    const int wave   = tid >> 5;
    const int lane   = tid & 31;
    const int mrow   = lane & 15;   // M-row (A) / N-col (B) within tile
    const int khalf  = lane >> 4;   // which half of K this lane holds

    const float* Pb = P + (size_t)b * N_PTS * 3;
    const float* Qb = Q + (size_t)b * N_PTS * 3;

    // ---- A-matrix fragment: row = this wave's 16 pred points -------------
    // Layout (16x4 f32 A): lanes 0-15 hold K=0,1 in v0,v1; lanes 16-31 K=2,3.
    // A row = [x, y, z, |p|^2]  so  A.B = |p|^2 - 2 p.q
    const int grow = rowblk * ROWS_PER_WG + wave * 16 + mrow;
    const float px = Pb[grow * 3 + 0];
    const float py = Pb[grow * 3 + 1];
    const float pz = Pb[grow * 3 + 2];
    const float pn = px * px + py * py + pz * pz;
    v2f a;
    a.x = khalf ? pz : px;
    a.y = khalf ? pn : py;

    v8f rowmin;
#pragma unroll
    for (int r = 0; r < 8; ++r) rowmin[r] = 3.0e38f;

    for (int c0 = 0; c0 < N_PTS; c0 += CHUNK) {
        __syncthreads();  // previous chunk fully consumed
        // Cooperative fill: B-form + norms for this chunk of target points.
        for (int i = tid; i < CHUNK; i += 256) {
            const float qx = Qb[(size_t)(c0 + i) * 3 + 0];
            const float qy = Qb[(size_t)(c0 + i) * 3 + 1];
            const float qz = Qb[(size_t)(c0 + i) * 3 + 2];
            sB[i] = make_float4(-2.0f * qx, -2.0f * qy, -2.0f * qz, 1.0f);
            sN[i] = qx * qx + qy * qy + qz * qz;
        }
        __syncthreads();

        const float2* sB2 = (const float2*)sB;
#pragma unroll 4
        for (int t = 0; t < TILES_PER_CHUNK; ++t) {
            // B-matrix fragment (4x16 f32): lane holds column N = mrow,
            // K-half selected by khalf -> contiguous float2 in the float4.
            const int col = t * 16 + mrow;
            const float2 bv = sB2[col * 2 + khalf];  // ds_load_b64
            const float  qn = sN[col];               // ds_load_b32
            v2f bb; bb.x = bv.x; bb.y = bv.y;

            v8f c = {};  // zero accumulator
            v8f d = __builtin_amdgcn_wmma_f32_16x16x4_f32(
                false, a, false, bb, (short)0, c, false, false);

            // d2 = max(d + |q|^2, 0); rowmin = min(rowmin, d2)
            // == med3(0, rowmin, d + qn) since rowmin >= 0.
#pragma unroll
            for (int r = 0; r < 8; ++r)
                rowmin[r] = __builtin_amdgcn_fmed3f(0.0f, rowmin[r], d[r] + qn);
        }
    }

    // ---- Row-min reduction -----------------------------------------------
    // D layout: VGPR r = row r on lanes 0-15, row r+8 on lanes 16-31.
    // Butterfly-min within each 16-lane half.
#pragma unroll
    for (int r = 0; r < 8; ++r) {
        float v = rowmin[r];
        v = fminf(v, __shfl_xor(v, 1, 32));
        v = fminf(v, __shfl_xor(v, 2, 32));
        v = fminf(v, __shfl_xor(v, 4, 32));
        v = fminf(v, __shfl_xor(v, 8, 32));
        rowmin[r] = v;
    }
    if (mrow == 0) {  // lane 0 (rows 0..7) and lane 16 (rows 8..15)
        float s = 0.0f;
#pragma unroll
        for (int r = 0; r < 8; ++r) s += rowmin[r];
        atomicAdd(out, s * scale);
    }
}

extern "C" void kernel_launch(void* const* d_in, const int* in_sizes, int n_in,
                              void* d_out, int out_size, void* d_ws, size_t ws_size,
                              hipStream_t stream) {
    const float* pred   = (const float*)d_in[0];
    const float* target = (const float*)d_in[1];
    float* out = (float*)d_out;

    chamfer_init_kernel<<<1, 1, 0, stream>>>(out);

    // chamfer.mean() = (sum of all forward row-mins + all backward row-mins) / (B*N)
    const float scale = 1.0f / ((float)BATCHES * (float)N_PTS);
    dim3 grid(BATCHES * (N_PTS / ROWS_PER_WG));
    chamfer_dir_kernel<<<grid, 256, 0, stream>>>(pred, target, out, scale);
    chamfer_dir_kernel<<<grid, 256, 0, stream>>>(target, pred, out, scale);
}